// Attention_Module_68882685493549
// MI455X (gfx1250) — compile-verified
//
#include <hip/hip_runtime.h>
#include <math.h>

#define NPTS         131072
#define IMG_H        512
#define IMG_W        512
#define NPLANES      256                 // 4 * 64
#define PLANE_STRIDE (IMG_H * IMG_W)     // 262144 floats per (b,c) plane

typedef float v2f __attribute__((ext_vector_type(2)));
typedef float v8f __attribute__((ext_vector_type(8)));

// ---------------------------------------------------------------------------
// Kernel 1: per-point gate  a[n] = s/s,  s = sigmoid(x*w0 + y*w1) * att[n]
// Dot products computed with V_WMMA_F32_16X16X4_F32 (16 points per wave).
//   A (16x4 f32): lanes 0-15 hold K0(=x),K1(=y) in VGPR0,VGPR1;
//                 lanes 16-31 hold K2,K3 -> zeroed.
//   B (4x16 f32): rows K0=w0, K1=w1 broadcast over N; rows K2,K3 = 0.
//   D (16x16)   : every column identical; VGPR i, lane 0  -> point base+i,
//                                         VGPR i, lane 16 -> point base+8+i.
// ---------------------------------------------------------------------------
__global__ __launch_bounds__(256) void attn_gate_wmma_kernel(
    const float* __restrict__ pts,   // [N,2]
    const float* __restrict__ fc_w,  // [1,2]
    const float* __restrict__ att,   // [N,1]
    float* __restrict__ a_out)       // [N]
{
  const int lane = threadIdx.x & 31;
  const int wave = threadIdx.x >> 5;
  const int base = blockIdx.x * 128 + wave * 16;   // 16 points per wave
  const int m    = base + (lane & 15);

  // Unconditional loads (no EXEC divergence before WMMA), select via VALU.
  const float x  = pts[2 * m + 0];
  const float y  = pts[2 * m + 1];
  const float w0 = fc_w[0];
  const float w1 = fc_w[1];
  const bool  lo = (lane < 16);

  v2f A, B;
  A.x = lo ? x  : 0.0f;   // K0 rows (lanes 0-15), K2 zeroed (lanes 16-31)
  A.y = lo ? y  : 0.0f;   // K1 / K3
  B.x = lo ? w0 : 0.0f;
  B.y = lo ? w1 : 0.0f;
  v8f C = {};

  v8f D = __builtin_amdgcn_wmma_f32_16x16x4_f32(
      /*neg_a=*/false, A, /*neg_b=*/false, B,
      /*c_mod=*/(short)0, C, /*reuse_a=*/false, /*reuse_b=*/false);

  if (lane == 0 || lane == 16) {
    const int mb = base + (lane == 0 ? 0 : 8);
#pragma unroll
    for (int i = 0; i < 8; ++i) {
      float dot = D[i];
      float s   = 1.0f / (1.0f + __expf(-dot));   // sigmoid
      s        *= att[mb + i];
      a_out[mb + i] = s / s;   // faithful normalization over singleton axis
    }
  }
}

// ---------------------------------------------------------------------------
// Kernel 2: bilinear gather over all 256 planes per point.
// One thread per point; synchronized plane loop keeps the live image footprint
// to a few 1 MB planes (L2-resident despite ~32x request amplification).
// NT stores keep the 134 MB output stream out of L2; software prefetch pulls
// plane c+2's two cachelines (fixed +1 MB stride from a random base).
// ---------------------------------------------------------------------------
__global__ __launch_bounds__(256) void bilinear_attend_kernel(
    const float* __restrict__ pts,     // [N,2]
    const float* __restrict__ img,     // [4,64,512,512]
    const float* __restrict__ a_vals,  // [N] (kernel 1 output)
    float* __restrict__ out)           // [4,64,N]
{
  const int n = blockIdx.x * blockDim.x + threadIdx.x;

  const float x   = pts[2 * n + 0];
  const float y   = pts[2 * n + 1];
  const float x0f = floorf(x);
  const float y0f = floorf(y);
  const int   x0  = (int)x0f;
  const int   y0  = (int)y0f;
  const float fx  = x - x0f;
  const float fy  = y - y0f;
  const float a   = a_vals[n];

  // Fold the gate into the bilinear weights (1 mul saved * 33.5M outputs).
  const float tl = (1.0f - fx) * (1.0f - fy) * a;
  const float tr = fx * (1.0f - fy) * a;
  const float bl = (1.0f - fx) * fy * a;
  const float br = fx * fy * a;

  const float* p = img + (y0 * IMG_W + x0);

  // Warm the pipeline: prefetch the first planes' two rows.
  __builtin_prefetch(p, 0, 1);
  __builtin_prefetch(p + IMG_W, 0, 1);
  __builtin_prefetch(p + PLANE_STRIDE, 0, 1);
  __builtin_prefetch(p + PLANE_STRIDE + IMG_W, 0, 1);

#pragma unroll 4
  for (int c = 0; c < NPLANES; ++c) {
    const float* q  = p + c * PLANE_STRIDE;
    const int    cp = (c + 2 < NPLANES) ? (c + 2) : (NPLANES - 1);
    const float* qp = p + cp * PLANE_STRIDE;
    __builtin_prefetch(qp, 0, 1);                 // plane c+2, row y0
    __builtin_prefetch(qp + IMG_W, 0, 1);         // plane c+2, row y1

    // 4x b32 loads (x0 parity unknown -> cannot legally use b64 pairs).
    const float v00 = q[0];
    const float v01 = q[1];
    const float v10 = q[IMG_W];
    const float v11 = q[IMG_W + 1];

    const float val = tl * v00 + tr * v01 + bl * v10 + br * v11;
    __builtin_nontemporal_store(val, &out[c * NPTS + n]);
  }
}

// ---------------------------------------------------------------------------
extern "C" void kernel_launch(void* const* d_in, const int* in_sizes, int n_in,
                              void* d_out, int out_size, void* d_ws, size_t ws_size,
                              hipStream_t stream) {
  const float* pts  = (const float*)d_in[0];  // lidar_points [N,2]
  const float* img  = (const float*)d_in[1];  // original_img [4,64,512,512]
  const float* fc_w = (const float*)d_in[2];  // fc_w [1,2]
  const float* att  = (const float*)d_in[3];  // att_param [N,1]

  float* a_out    = (float*)d_out;            // first N floats: a
  float* attended = (float*)d_out + NPTS;     // then [4,64,N]

  // Kernel 1: 16 points / wave, 8 waves / block -> 128 points / block.
  attn_gate_wmma_kernel<<<NPTS / 128, 256, 0, stream>>>(pts, fc_w, att, a_out);

  // Kernel 2: one thread per point, 512 blocks x 256 threads.
  bilinear_attend_kernel<<<NPTS / 256, 256, 0, stream>>>(pts, img, a_out, attended);
}